// PGN_7078106104066
// MI455X (gfx1250) — compile-verified
//
#include <hip/hip_runtime.h>
#include <stdint.h>

// Problem constants (from the reference)
#define BATCH   16
#define DEC_LEN 50
#define ENC_LEN 400
#define VOCAB   50000
#define OOV_LEN 30
#define OUTV    (VOCAB + OOV_LEN)   // 50030 floats per output row
#define LDS_FLOATS 50032            // acc padded to multiple of 4 for float4 zero-fill
#define NTHR    1024                // compile-time block size (32 wave32 waves)

typedef __attribute__((ext_vector_type(4))) unsigned int u32x4;
typedef __attribute__((ext_vector_type(8))) int          i32x8;
typedef __attribute__((ext_vector_type(4))) int          i32x4;

#if __has_builtin(__builtin_amdgcn_tensor_store_from_lds)
#define HAVE_TDM 1
#else
#define HAVE_TDM 0
#endif

#if __has_builtin(__builtin_amdgcn_global_load_async_to_lds_b32) && \
    __has_builtin(__builtin_amdgcn_s_wait_asynccnt)
#define HAVE_ASYNC 1
#else
#define HAVE_ASYNC 0
#endif

#if HAVE_ASYNC
// clang says the builtin takes (addrspace(1) int*, addrspace(3) int*, Ii, Ii)
typedef __attribute__((address_space(1))) int* gint_t;
typedef __attribute__((address_space(3))) int* lint_t;
__device__ __forceinline__ gint_t to_global(const void* p) {
    return (gint_t)(uintptr_t)p;            // same 64-bit VA in AS1
}
__device__ __forceinline__ lint_t to_lds(const void* p) {
    return (lint_t)(uint32_t)(uintptr_t)p;  // low 32 bits of generic = LDS offset
}
#endif

__global__ __launch_bounds__(NTHR)
void pgn_final_dist_kernel(const int*   __restrict__ enc_idx,     // [B, ENC_LEN]
                           const float* __restrict__ vocab_dists, // [B, T, VOCAB]
                           const float* __restrict__ attn_dists,  // [B, T, ENC_LEN]
                           const float* __restrict__ p_gens,      // [B, T, 1]
                           float*       __restrict__ out)         // [B, T, OUTV]
{
    extern __shared__ float acc[];            // LDS_FLOATS floats + 800-float staging area
    int*   sidx  = reinterpret_cast<int*>(acc + LDS_FLOATS);        // [ENC_LEN]
    float* sattn = acc + LDS_FLOATS + ENC_LEN;                       // [ENC_LEN]

    const int row = blockIdx.x;               // row = b*DEC_LEN + t, 0..799
    const int b   = row / DEC_LEN;            // const divide -> mulhi, scalar
    const int tid = threadIdx.x;

#if HAVE_ASYNC
    // Kick off ASYNCcnt DMA of the scatter indices + attention row into LDS
    // staging; this overlaps with the 200KB LDS zero-fill below.
    if (tid < ENC_LEN) {
        __builtin_amdgcn_global_load_async_to_lds_b32(
            to_global(enc_idx + b * ENC_LEN + tid),              to_lds(&sidx[tid]),  0, 0);
        __builtin_amdgcn_global_load_async_to_lds_b32(
            to_global(attn_dists + (size_t)row * ENC_LEN + tid), to_lds(&sattn[tid]), 0, 0);
    }
#endif

    // ---------- phase 0: zero the LDS row accumulator (ds_store_b128) ----------
    float4* acc4 = reinterpret_cast<float4*>(acc);
    #pragma unroll 4
    for (int i = tid; i < LDS_FLOATS / 4; i += NTHR)
        acc4[i] = make_float4(0.f, 0.f, 0.f, 0.f);

    const float pg = p_gens[row];             // uniform per block

    __syncthreads();

    // ---------- phase 1: scatter attn*(1-pg) into LDS (ds_add_f32, no-return) ----------
    // Adds commute with phase 2's adds, so no barrier between the two phases.
    if (tid < ENC_LEN) {
#if HAVE_ASYNC
        __builtin_amdgcn_s_wait_asynccnt(0);  // lane reads the element its own wave fetched
        const int   v = sidx[tid];
        const float a = sattn[tid] * (1.0f - pg);
#else
        const int   v = enc_idx[b * ENC_LEN + tid];            // 0 <= v < OUTV
        const float a = attn_dists[(size_t)row * ENC_LEN + tid] * (1.0f - pg);
#endif
        atomicAdd(&acc[v], a);                                 // ds_add_f32
    }

    // ---------- phase 2: dense vocab*pg accumulated into LDS ----------
    // vocab rows are 16B aligned (row * 200000 bytes) -> global_load_b128 stream.
    const float4* vrow = reinterpret_cast<const float4*>(vocab_dists + (size_t)row * VOCAB);
    #pragma unroll 2
    for (int i = tid; i < VOCAB / 4; i += NTHR) {
        const float4 v = vrow[i];
        atomicAdd(&acc[4 * i + 0], v.x * pg);
        atomicAdd(&acc[4 * i + 1], v.y * pg);
        atomicAdd(&acc[4 * i + 2], v.z * pg);
        atomicAdd(&acc[4 * i + 3], v.w * pg);
    }

    __syncthreads();   // s_wait_dscnt 0 + barrier: accumulator complete

    // ---------- phase 3: DMA the finished 200KB row from LDS to HBM (TDM) ----------
#if HAVE_TDM
    if (tid == 0) {   // single TDM issue from wave 0 (EXEC ignored by TENSOR ops)
        const uint64_t ga       = (uint64_t)(uintptr_t)(out + (size_t)row * OUTV);
        const uint32_t lds_base = (uint32_t)(uintptr_t)acc;

        // ---- D# group 0 (128b): count=1 | lds_addr | global_addr[56:0] | type=2 ----
        u32x4 g0;
        g0.x = 1u;                                           // count=1, user descriptor
        g0.y = lds_base;                                     // lds_addr (bytes)
        g0.z = (uint32_t)(ga & 0xFFFFFFFFu);                 // global_addr[31:0]
        g0.w = (uint32_t)((ga >> 32) & 0x1FFFFFFu)           // global_addr[56:32]
             | 0x80000000u;                                  // type = 2 ("image")

        // ---- D# group 1 (256b): 1-D tensor, data_size=4B, dim0=tile0=50030 ----
        i32x8 g1;
        g1[0] = (int)(2u << 16);                             // workgroup_mask=0, data_size=2 (4B)
        g1[1] = (int)((uint32_t)(OUTV & 0xFFFF) << 16);      // tensor_dim0[15:0] @ bits[63:48]
        g1[2] = (int)(1u << 16);                             // tensor_dim0[31:16]=0, tensor_dim1=1
        g1[3] = (int)((uint32_t)OUTV << 16);                 // tile_dim0 = 50030 @ bits[127:112]
        g1[4] = 1;                                           // tile_dim1 = 1, tile_dim2 = 0
        g1[5] = OUTV;                                        // tensor_dim0_stride[31:0]
        g1[6] = 0;
        g1[7] = 0;

        i32x4 z4 = {0, 0, 0, 0};                             // groups 2/3 unused (<=2D)
  #if __has_include(<hip/amd_detail/amd_gfx1250_TDM.h>)
        i32x8 z8 = {0, 0, 0, 0, 0, 0, 0, 0};                 // therock clang-23: 6-arg form
        __builtin_amdgcn_tensor_store_from_lds(g0, g1, z4, z4, z8, 0);
  #else
        __builtin_amdgcn_tensor_store_from_lds(g0, g1, z4, z4, 0); // ROCm 7.2: 5-arg form
  #endif
        __builtin_amdgcn_s_wait_tensorcnt(0);                // drain TENSORcnt before wave exit
    }
#else
    // Fallback: vectorized LDS->global streaming store.
    // Output rows are only 8B aligned (50030*4 % 16 == 8), so float2 stores.
    const float2* accs2 = reinterpret_cast<const float2*>(acc);
    float2* orow = reinterpret_cast<float2*>(out + (size_t)row * OUTV);
    #pragma unroll 4
    for (int i = tid; i < OUTV / 2; i += NTHR)
        orow[i] = accs2[i];
#endif
}

extern "C" void kernel_launch(void* const* d_in, const int* in_sizes, int n_in,
                              void* d_out, int out_size, void* d_ws, size_t ws_size,
                              hipStream_t stream) {
    const int*   enc = (const int*)  d_in[0];  // enc_extended_inp, int32
    const float* vd  = (const float*)d_in[1];  // vocab_dists
    const float* ad  = (const float*)d_in[2];  // attn_dists
    const float* pg  = (const float*)d_in[3];  // p_gens
    // d_in[4] = batch_oov_len (compile-time 30 here)
    float* out = (float*)d_out;

    const dim3 grid(BATCH * DEC_LEN);          // 800 rows, one WG per (b,t)
    const dim3 block(NTHR);                    // 32 wave32 waves
    const size_t shmem = (LDS_FLOATS + 2 * ENC_LEN) * sizeof(float);  // 203328 B of LDS

    hipLaunchKernelGGL(pgn_final_dist_kernel, grid, block, shmem, stream,
                       enc, vd, ad, pg, out);
}